// AttentionLayer_61795989454868
// MI455X (gfx1250) — compile-verified
//
#include <hip/hip_runtime.h>
#include <math.h>
#include <stdint.h>

typedef float v2f __attribute__((ext_vector_type(2)));
typedef float v4f __attribute__((ext_vector_type(4)));
typedef float v8f __attribute__((ext_vector_type(8)));
typedef unsigned int u32x4 __attribute__((ext_vector_type(4)));
typedef int i32x4 __attribute__((ext_vector_type(4)));
typedef int i32x8 __attribute__((ext_vector_type(8)));

#define B_SZ 2048
#define N_SZ 64
#define D_SZ 256
#define H_SZ 16
#define CH_K 128          // K columns per LDS chunk
#define A_STRIDE 132      // padded LDS row stride (floats); 132 % 64 = 4 -> conflict-free b64 A reads

#if __has_builtin(__builtin_amdgcn_tensor_load_to_lds) && __has_builtin(__builtin_amdgcn_s_wait_tensorcnt)
#define USE_TDM 1
#else
#define USE_TDM 0
#endif

__global__ __launch_bounds__(128) void attn_wmma_kernel(
    const float* __restrict__ members,  // [B, N, D]
    const float* __restrict__ item,     // [B, D]
    const float* __restrict__ W1,       // [D, H]
    const float* __restrict__ b1,       // [H]
    const float* __restrict__ W2,       // [H] (H,1)
    const float* __restrict__ b2,       // [1]
    float* __restrict__ out)            // [B, N]
{
    __shared__ float Asmem[N_SZ * A_STRIDE];   // 64 x 132 member tile (33.8 KB)
    __shared__ float W1s[D_SZ * H_SZ];         // item-scaled W1, 256x16 (16 KB)
    __shared__ float Gs[N_SZ * H_SZ];          // G = U @ W1s, 64x16
    __shared__ float ts[H_SZ];                 // column sums of G
    __shared__ float ls[N_SZ];                 // logits
    __shared__ float es[N_SZ];                 // exp(logit - max)

    const int b    = blockIdx.x;
    const int tid  = threadIdx.x;
    const int lane = tid & 31;
    const int wv   = tid >> 5;     // wave id 0..3 -> owns G rows [16wv, 16wv+16)
    const int half = lane >> 4;    // 0: K=0,1 half / 1: K=2,3 half (A layout)
    const int l16  = lane & 15;

    // ---- stage 0: W1s[d,h] = item[b,d] * W1[d,h] (vectorized float4) ----
    const float* itemb = item + b * D_SZ;
    {
        const v4f* W1v = (const v4f*)W1;
        #pragma unroll
        for (int i = 0; i < 8; ++i) {
            int vi   = i * 128 + tid;        // 1024 float4 total
            v4f w    = W1v[vi];
            int base = vi * 4;               // base%16 in {0,4,8,12}: all 4 elems same d
            float s  = itemb[base >> 4];
            *(v4f*)&W1s[base] = w * s;
        }
    }

    const float* memb = members + (size_t)b * (N_SZ * D_SZ);
    v8f c = {0.f, 0.f, 0.f, 0.f, 0.f, 0.f, 0.f, 0.f};

    // ---- stream members through LDS in two 64x128 chunks, WMMA-accumulate ----
    for (int ch = 0; ch < 2; ++ch) {
        __syncthreads();   // previous chunk fully consumed before overwrite

#if USE_TDM
        // Tensor Data Mover: DMA 64x128 f32 tile (row stride 256 floats in memory)
        // into LDS with hardware padding: pad_interval=6 (every 128 DWORDs),
        // pad_amount=3 (4 DWORDs) -> LDS row stride 132 floats (conflict-free).
        if (__builtin_amdgcn_readfirstlane(tid) == 0) {   // scalar branch: wave 0 only
            uint64_t ga = (uint64_t)(uintptr_t)(memb + ch * CH_K);   // tile start
            uint32_t la = (uint32_t)(uintptr_t)&Asmem[0];            // LDS byte offset

            u32x4 g0;
            g0.x = 1u;                                    // count=1 valid descriptor
            g0.y = la;                                    // lds_addr [63:32]
            g0.z = (uint32_t)ga;                          // global_addr [95:64]
            g0.w = (uint32_t)(ga >> 32) | (2u << 30);     // addr[56:32] | type=2

            i32x8 g1;
            g1[0] = (int)((2u << 16)        // data_size = 4B
                        | (1u << 20)        // pad_enable
                        | (6u << 22)        // pad_interval: 128 DWORDs
                        | (3u << 25));      // pad_amount: 4 DWORDs
            g1[1] = (int)(256u << 16);      // tensor_dim0 = 256 (low 16 @ bits 63:48)
            g1[2] = (int)(64u << 16);       // tensor_dim0 hi=0 | tensor_dim1 = 64
            g1[3] = (int)(128u << 16);      // tensor_dim1 hi=0 | tile_dim0 = 128
            g1[4] = 64;                     // tile_dim1 = 64 | tile_dim2 = 0
            g1[5] = 256;                    // tensor_dim0_stride = 256 (low 32)
            g1[6] = 0;                      // stride hi | tensor_dim1_stride lo
            g1[7] = 0;

            i32x4 gz4 = {0, 0, 0, 0};            // groups 2/3 unused (2-D tensor)
            i32x8 gz8 = {0, 0, 0, 0, 0, 0, 0, 0};
            __builtin_amdgcn_tensor_load_to_lds(g0, g1, gz4, gz4, gz8, 0);
            __builtin_amdgcn_s_wait_tensorcnt(0);
        }
#else
        #pragma unroll
        for (int i = 0; i < 16; ++i) {
            int vi = i * 128 + tid;          // 2048 float4 = 64x128 floats
            int n  = vi >> 5;                // 32 float4 per row-chunk
            int cw = (vi & 31) << 2;         // col within chunk
            v4f v = *(const v4f*)&memb[n * D_SZ + ch * CH_K + cw];
            *(v4f*)&Asmem[n * A_STRIDE + cw] = v;
        }
#endif
        __syncthreads();   // tile ready in LDS for all waves

        const int arow = (wv * 16 + l16) * A_STRIDE;
        #pragma unroll
        for (int kk = 0; kk < 32; ++kk) {
            // A 16x4 f32 fragment: lanes 0-15 hold K=0,1; lanes 16-31 hold K=2,3
            v2f a = *(const v2f*)&Asmem[arow + kk * 4 + 2 * half];
            // B 4x16 f32 fragment: VGPR0 rows K=0(1), VGPR1 rows K=2(3) per half-wave
            int krow = (ch * CH_K + kk * 4 + half) * H_SZ + l16;
            v2f bb;
            bb.x = W1s[krow];
            bb.y = W1s[krow + 2 * H_SZ];
            c = __builtin_amdgcn_wmma_f32_16x16x4_f32(
                    false, a, false, bb, (short)0, c, false, false);
        }
    }

    // ---- spill G tile: C layout VGPR j -> row j (lanes 0-15) / row j+8 (lanes 16-31) ----
    #pragma unroll
    for (int j = 0; j < 8; ++j) {
        int row = wv * 16 + j + 8 * half;
        Gs[row * H_SZ + l16] = c[j];
    }
    __syncthreads();

    // ---- t[h] = sum_n G[n,h] ----
    if (tid < H_SZ) {
        float s = 0.f;
        for (int n = 0; n < N_SZ; ++n) s += Gs[n * H_SZ + tid];
        ts[tid] = s;
    }
    __syncthreads();

    // ---- logits: relu(64*G - t + b1) . W2 + b2 ----
    if (tid < N_SZ) {
        float acc = b2[0];
        #pragma unroll
        for (int h = 0; h < H_SZ; ++h) {
            float v = 64.f * Gs[tid * H_SZ + h] - ts[h] + b1[h];
            v = v > 0.f ? v : 0.f;
            acc = fmaf(v, W2[h], acc);
        }
        ls[tid] = acc;
    }
    __syncthreads();

    // ---- softmax over the 64 members ----
    float e = 0.f;
    if (tid < N_SZ) {
        float m = -INFINITY;
        for (int n = 0; n < N_SZ; ++n) m = fmaxf(m, ls[n]);
        e = __expf(ls[tid] - m);
        es[tid] = e;
    }
    __syncthreads();
    if (tid < N_SZ) {
        float s = 0.f;
        for (int n = 0; n < N_SZ; ++n) s += es[n];
        out[b * N_SZ + tid] = e / s;
    }
}

extern "C" void kernel_launch(void* const* d_in, const int* in_sizes, int n_in,
                              void* d_out, int out_size, void* d_ws, size_t ws_size,
                              hipStream_t stream) {
    const float* members = (const float*)d_in[0];  // [2048,64,256]
    const float* item    = (const float*)d_in[1];  // [2048,256]
    const float* W1      = (const float*)d_in[2];  // [256,16]
    const float* b1      = (const float*)d_in[3];  // [16]
    const float* W2      = (const float*)d_in[4];  // [16,1]
    const float* b2      = (const float*)d_in[5];  // [1]
    float* out           = (float*)d_out;          // [2048,64]

    attn_wmma_kernel<<<dim3(B_SZ), dim3(128), 0, stream>>>(
        members, item, W1, b1, W2, b2, out);
}